// Gev_45612552684113
// MI455X (gfx1250) — compile-verified
//
#include <hip/hip_runtime.h>
#include <stdint.h>

#define TPB 128          // 4 wave32s per block; 2 x 24 KB LDS buffers
#define MAXBLK 2048      // persistent grid size (grid-stride over tiles)

// ---------------- complex helpers (register-resident, fmaf-based) ----------
struct cx { float r, i; };
__device__ __forceinline__ cx cmk(float r, float i){ cx c; c.r = r; c.i = i; return c; }
__device__ __forceinline__ cx cscale(cx a, float s){ return cmk(a.r * s, a.i * s); }
__device__ __forceinline__ cx csub(cx a, cx b){ return cmk(a.r - b.r, a.i - b.i); }
// acc += a*b
__device__ __forceinline__ cx cfma(cx a, cx b, cx c){
  c.r = fmaf(a.r, b.r, fmaf(-a.i, b.i, c.r));
  c.i = fmaf(a.r, b.i, fmaf( a.i, b.r, c.i));
  return c;
}
// acc += a*conj(b)
__device__ __forceinline__ cx cfmac(cx a, cx b, cx c){
  c.r = fmaf(a.r, b.r, fmaf( a.i, b.i, c.r));
  c.i = fmaf(a.i, b.r, fmaf(-a.r, b.i, c.i));
  return c;
}

// ---------------- CDNA5 async global->LDS (ASYNCcnt path, GVS mode) --------
__device__ __forceinline__ uint32_t lds_off(const void* p){
  // flat LDS address: addr[31:0] is the wave-relative LDS byte offset (ISA 10.2)
  return (uint32_t)(uintptr_t)p;
}
// saddr (SGPR base) + 32-bit per-lane byte offset + shared IOFFSET immediate;
// IOFFSET is applied to BOTH the global and LDS addresses (ISA 08 4.4).
#define ASYNC_B128_OFF(ldsa, voff, gbase, OFF)                                 \
  asm volatile("global_load_async_to_lds_b128 %0, %1, %2 offset:" #OFF        \
               " th:TH_LOAD_NT"                                                \
               :: "v"(ldsa), "v"(voff), "s"(gbase) : "memory")
__device__ __forceinline__ void wait_async0(){
  asm volatile("s_wait_asynccnt 0" ::: "memory");
}

// ---- stage one block-tile (SS 80B + NN 80B + X 32B per lane) into LDS -----
__device__ __forceinline__ void issue_tile_loads(
    unsigned long long gx, unsigned long long gss, unsigned long long gnn,
    float* buf, int tile, int N, int tid)
{
  const int p = tile * TPB + tid;
  if (p >= N) return;
  const uint32_t voffm = (uint32_t)p * 80u;   // SS/NN byte offset (fits 32-bit)
  const uint32_t voffx = (uint32_t)p * 32u;
  const uint32_t lss = lds_off(buf + tid * 20);
  const uint32_t lnn = lds_off(buf + TPB * 20 + tid * 20);
  const uint32_t lx  = lds_off(buf + TPB * 40 + tid * 8);
  ASYNC_B128_OFF(lss, voffm, gss, 0);
  ASYNC_B128_OFF(lss, voffm, gss, 16);
  ASYNC_B128_OFF(lss, voffm, gss, 32);
  ASYNC_B128_OFF(lss, voffm, gss, 48);
  ASYNC_B128_OFF(lss, voffm, gss, 64);
  ASYNC_B128_OFF(lnn, voffm, gnn, 0);
  ASYNC_B128_OFF(lnn, voffm, gnn, 16);
  ASYNC_B128_OFF(lnn, voffm, gnn, 32);
  ASYNC_B128_OFF(lnn, voffm, gnn, 48);
  ASYNC_B128_OFF(lnn, voffm, gnn, 64);
  ASYNC_B128_OFF(lx,  voffx, gx, 0);
  ASYNC_B128_OFF(lx,  voffx, gx, 16);
}

// ---- full per-problem GEV pipeline (all registers) ------------------------
__device__ __forceinline__ void compute_one(
    const float* __restrict__ s, const float* __restrict__ n,
    const float* __restrict__ x, float* __restrict__ out, int p)
{
  // rebuild Hermitian SS and diagonally-loaded NN (triu order, C=4)
  const int IUa[10] = {0,0,0,0,1,1,1,2,2,3};
  const int JUa[10] = {0,1,2,3,1,2,3,2,3,3};
  cx S[4][4], A[4][4];
#pragma unroll
  for (int q = 0; q < 10; ++q) {
    const int i = IUa[q], j = JUa[q];
    const float sr = s[q], si = s[10 + q];
    const float nr = n[q], ni = n[10 + q];
    if (i == j) { S[i][i] = cmk(sr, 0.f); A[i][i] = cmk(nr, 0.f); }
    else { S[i][j] = cmk(sr,  si); S[j][i] = cmk(sr, -si);
           A[i][j] = cmk(nr,  ni); A[j][i] = cmk(nr, -ni); }
  }
  const float lam = 1e-3f * (n[0] + n[4] + n[7] + n[9]) + 1e-20f;
#pragma unroll
  for (int i = 0; i < 4; ++i) A[i][i].r += lam;

  // Cholesky NN = L L^H (lower, real diagonal; real-only divides)
  cx L[4][4];
#pragma unroll
  for (int k = 0; k < 4; ++k) {
    const float d  = fmaxf(A[k][k].r, 1e-30f);
    const float sd = sqrtf(d);
    const float iv = 1.0f / sd;
    L[k][k] = cmk(sd, 0.f);
#pragma unroll
    for (int i = k + 1; i < 4; ++i) L[i][k] = cscale(A[i][k], iv);
#pragma unroll
    for (int i = k + 1; i < 4; ++i)
#pragma unroll
      for (int j = k + 1; j <= i; ++j)
        A[i][j] = csub(A[i][j], cfmac(L[i][k], L[j][k], cmk(0.f, 0.f)));
  }

  // Li = L^-1 (lower triangular)
  cx Li[4][4];
#pragma unroll
  for (int k = 0; k < 4; ++k) {
    Li[k][k] = cmk(1.0f / L[k][k].r, 0.f);
#pragma unroll
    for (int i = k + 1; i < 4; ++i) {
      cx acc = cmk(0.f, 0.f);
#pragma unroll
      for (int m = k; m < i; ++m) acc = cfma(L[i][m], Li[m][k], acc);
      const float di = -1.0f / L[i][i].r;
      Li[i][k] = cscale(acc, di);
    }
  }

  // Cm = Li * SS * Li^H : Hermitian -> compute upper triangle only, mirror
  cx Cm[4][4];
#pragma unroll
  for (int i = 0; i < 4; ++i) {
    cx Trow[4];
#pragma unroll
    for (int j = 0; j < 4; ++j) {
      cx acc = cmk(0.f, 0.f);
#pragma unroll
      for (int m = 0; m <= i; ++m) acc = cfma(Li[i][m], S[m][j], acc);
      Trow[j] = acc;
    }
#pragma unroll
    for (int j = i; j < 4; ++j) {
      cx acc = cmk(0.f, 0.f);
#pragma unroll
      for (int m = 0; m <= j; ++m) acc = cfmac(Trow[m], Li[j][m], acc);
      Cm[i][j] = acc;
    }
  }
#pragma unroll
  for (int i = 0; i < 4; ++i) {
    Cm[i][i].i = 0.f;                                  // exact-real diagonal
#pragma unroll
    for (int j = i + 1; j < 4; ++j) Cm[j][i] = cmk(Cm[i][j].r, -Cm[i][j].i);
  }

  // principal eigenvector: Cm^32 by 5 Hermitian squarings (+trace renorm)
  cx M[4][4];
  {
    const float tr  = Cm[0][0].r + Cm[1][1].r + Cm[2][2].r + Cm[3][3].r;
    const float isc = 1.0f / (tr + 1e-30f);
#pragma unroll
    for (int i = 0; i < 4; ++i)
#pragma unroll
      for (int j = 0; j < 4; ++j) M[i][j] = cscale(Cm[i][j], isc);
  }
#pragma unroll 1
  for (int it = 0; it < 5; ++it) {
    float dg[4];                                       // (M^2)_ii = sum |M_im|^2
#pragma unroll
    for (int i = 0; i < 4; ++i) {
      float a = 0.f;
#pragma unroll
      for (int m = 0; m < 4; ++m)
        a = fmaf(M[i][m].r, M[i][m].r, fmaf(M[i][m].i, M[i][m].i, a));
      dg[i] = a;
    }
    cx up[4][4];                                       // strict upper of M^2
#pragma unroll
    for (int i = 0; i < 4; ++i)
#pragma unroll
      for (int j = i + 1; j < 4; ++j) {
        cx acc = cmk(0.f, 0.f);
#pragma unroll
        for (int m = 0; m < 4; ++m) acc = cfma(M[i][m], M[m][j], acc);
        up[i][j] = acc;
      }
    const float tr  = dg[0] + dg[1] + dg[2] + dg[3];
    const float isc = 1.0f / (tr + 1e-30f);
#pragma unroll
    for (int i = 0; i < 4; ++i) {
      M[i][i] = cmk(dg[i] * isc, 0.f);
#pragma unroll
      for (int j = i + 1; j < 4; ++j) {
        M[i][j] = cscale(up[i][j], isc);
        M[j][i] = cmk(M[i][j].r, -M[i][j].i);
      }
    }
  }
  // pick column with largest diagonal (branchless v_cndmask selects)
  cx y[4] = { M[0][0], M[1][0], M[2][0], M[3][0] };
  float best = M[0][0].r;
#pragma unroll
  for (int j = 1; j < 4; ++j) {
    const bool b = M[j][j].r > best;
    best = b ? M[j][j].r : best;
#pragma unroll
    for (int i = 0; i < 4; ++i) y[i] = b ? M[i][j] : y[i];
  }
#pragma unroll
  for (int rep = 0; rep < 2; ++rep) {                  // two power refinements
    cx z[4];
#pragma unroll
    for (int i = 0; i < 4; ++i) {
      cx acc = cmk(0.f, 0.f);
#pragma unroll
      for (int m = 0; m < 4; ++m) acc = cfma(Cm[i][m], y[m], acc);
      z[i] = acc;
    }
#pragma unroll
    for (int i = 0; i < 4; ++i) y[i] = z[i];
  }

  // w = Li^H y, unit-normalize, beamform with X
  cx w[4];
#pragma unroll
  for (int i = 0; i < 4; ++i) {
    cx acc = cmk(0.f, 0.f);
#pragma unroll
    for (int m = i; m < 4; ++m) acc = cfmac(y[m], Li[m][i], acc);
    w[i] = acc;
  }
  float nrm = 0.f;
#pragma unroll
  for (int i = 0; i < 4; ++i) nrm = fmaf(w[i].r, w[i].r, fmaf(w[i].i, w[i].i, nrm));
  const float ivn = rsqrtf(fmaxf(nrm, 1e-38f));

  cx Y = cmk(0.f, 0.f);
#pragma unroll
  for (int c = 0; c < 4; ++c) {
    const cx xc = cmk(x[c], x[4 + c]);
    Y = cfma(cscale(w[c], ivn), xc, Y);
  }
  __builtin_nontemporal_store(Y.r, out + (size_t)p * 2u);
  __builtin_nontemporal_store(Y.i, out + (size_t)p * 2u + 1u);
}

// ---------------------------------------------------------------------------
__global__ __launch_bounds__(TPB) void gev_beamform_kernel(
    const float* __restrict__ Xs, const float* __restrict__ SSs,
    const float* __restrict__ NNs, float* __restrict__ out, int N)
{
  __shared__ float sbuf[2][TPB * 48];    // [SS 20 | NN 20 | X 8] floats/lane

  const int tid    = threadIdx.x;
  const int ntiles = (N + TPB - 1) / TPB;
  const unsigned long long gx  = (unsigned long long)(uintptr_t)Xs;
  const unsigned long long gss = (unsigned long long)(uintptr_t)SSs;
  const unsigned long long gnn = (unsigned long long)(uintptr_t)NNs;

  int tile = blockIdx.x;
  int cur  = 0;
  issue_tile_loads(gx, gss, gnn, &sbuf[0][0], tile, N, tid);   // prologue
  wait_async0();
  __syncthreads();

  for (; tile < ntiles; tile += gridDim.x) {
    const int nxt = tile + gridDim.x;
    if (nxt < ntiles)                                          // prefetch t+1
      issue_tile_loads(gx, gss, gnn, &sbuf[cur ^ 1][0], nxt, N, tid);

    const int p = tile * TPB + tid;
    if (p < N) {
      const float* base = &sbuf[cur][0];
      compute_one(base + tid * 20, base + TPB * 20 + tid * 20,
                  base + TPB * 40 + tid * 8, out, p);
    }
    wait_async0();       // prefetch landed (overlapped with ~1k VALU ops)
    __syncthreads();     // all waves done reading buf[cur] before reuse
    cur ^= 1;
  }
}

extern "C" void kernel_launch(void* const* d_in, const int* in_sizes, int n_in,
                              void* d_out, int out_size, void* d_ws, size_t ws_size,
                              hipStream_t stream) {
  (void)n_in; (void)out_size; (void)d_ws; (void)ws_size;
  const float* Xs  = (const float*)d_in[0];
  const float* SSs = (const float*)d_in[1];
  const float* NNs = (const float*)d_in[2];
  float* out = (float*)d_out;
  const int N = in_sizes[0] / 8;            // (B,T,F,2,C=4) -> 8 floats/problem
  const int ntiles = (N + TPB - 1) / TPB;
  const int blocks = ntiles < MAXBLK ? ntiles : MAXBLK;
  gev_beamform_kernel<<<blocks, TPB, 0, stream>>>(Xs, SSs, NNs, out, N);
}